// RGCN_33354716020861
// MI455X (gfx1250) — compile-verified
//
#include <hip/hip_runtime.h>
#include <hip/hip_bf16.h>

// MI455X / gfx1250 fused R-GCN layer: V_WMMA_F32_16X16X4_F32 + Tensor Data Mover.
// One workgroup (1024 threads = 32 wave32) per batch b.
//   Phase 1: x[e] = node@We[e]+be[e] (e=0..3) -> LDS; self = node@Wf+bf stays in VGPRs
//            (phase-1 tile (jt,ot) == phase-2 tile (it,lt), identical C/D layout).
//   Phase 2: agg = sum_k edge[..,k+1] @ x[..,k]; 32-column edge tiles streamed
//            global->LDS by TDM into double buffers (TENSORcnt-waited), accumulators
//            seeded with the in-register self_msg; relu fused on store.

typedef __attribute__((ext_vector_type(2))) float v2f;
typedef __attribute__((ext_vector_type(8))) float v8f;
typedef unsigned int u32x4 __attribute__((ext_vector_type(4)));
typedef int          i32x4 __attribute__((ext_vector_type(4)));
typedef int          i32x8 __attribute__((ext_vector_type(8)));

#define BN    512
#define NN    128
#define NIN_  64
#define NOUT_ 64
#define EC    5            // edge channels (channel 0 dropped)

#define NODE_LDA 68        // padded row stride (words): conflict-free phase-1 A fetch
#define JB       32        // j-columns per streamed block
#define NJB      (NN / JB) // 4
#define EB_W     (JB * EC) // 160 data words per row per block
#define EB_LDA   165       // 160 + 5 pads (TDM: 1 dword per 32) -> stride 37 mod 64, conflict-free

#if defined(__gfx1250__) && __has_builtin(__builtin_amdgcn_tensor_load_to_lds)
#define HAVE_TDM 1
#pragma message("CDNA5: tensor_load_to_lds builtin AVAILABLE -> TDM path compiled")
#else
#define HAVE_TDM 0
#pragma message("CDNA5: tensor_load_to_lds builtin MISSING -> synchronous fallback compiled")
#endif

#if HAVE_TDM
// One TDM DMA: 2D tile (EB_W x NN dwords) from global edge rows (stride 640 dwords)
// into LDS at lds_byte, padding 1 dword after every 32 dwords (row stride EB_LDA).
__device__ __forceinline__ void tdm_load_edge_block(const float* gsrc, unsigned lds_byte) {
    const unsigned long long ga = (unsigned long long)(uintptr_t)gsrc;
    u32x4 g0;
    g0[0] = 1u;                                   // count=1, user descriptor
    g0[1] = lds_byte;                             // lds_addr (bytes)
    g0[2] = (unsigned)ga;                         // global_addr[31:0]
    g0[3] = ((unsigned)(ga >> 32) & 0x01FFFFFFu)  // global_addr[56:32]
            | 0x80000000u;                        // type=2 ("image")
    i32x8 g1;
    g1[0] = (int)((2u << 16)                      // data_size = 4B
                | (1u << 20)                      // pad_enable
                | (4u << 22)                      // pad_interval = 32 dwords
                | (0u << 25));                    // pad_amount = 1 dword
    g1[1] = (int)(((unsigned)(NN * EC)) << 16);   // tensor_dim0 = 640 (low 16)
    g1[2] = (int)(((unsigned)NN) << 16);          // tensor_dim0 hi=0 | tensor_dim1 = 128
    g1[3] = (int)(((unsigned)EB_W) << 16);        // tensor_dim1 hi=0 | tile_dim0 = 160
    g1[4] = NN;                                   // tile_dim1 = 128, tile_dim2 = 0
    g1[5] = NN * EC;                              // tensor_dim0_stride = 640 (low 32)
    g1[6] = 0;                                    // stride hi | tensor_dim1_stride (unused, 2D)
    g1[7] = 0;
    i32x4 gz; gz[0] = 0; gz[1] = 0; gz[2] = 0; gz[3] = 0;
#if __clang_major__ >= 23
#pragma message("CDNA5: using 6-arg tensor_load_to_lds (clang>=23)")
    i32x8 gz8; gz8[0]=0; gz8[1]=0; gz8[2]=0; gz8[3]=0; gz8[4]=0; gz8[5]=0; gz8[6]=0; gz8[7]=0;
    __builtin_amdgcn_tensor_load_to_lds(g0, g1, gz, gz, gz8, 0);
#else
#pragma message("CDNA5: using 5-arg tensor_load_to_lds (clang-22 / ROCm 7.2)")
    __builtin_amdgcn_tensor_load_to_lds(g0, g1, gz, gz, 0);
#endif
}
#endif

__global__ __launch_bounds__(1024)
void rgcn_fused_wmma(const float* __restrict__ node,
                     const float* __restrict__ edge,
                     const float* __restrict__ We,
                     const float* __restrict__ be,
                     const float* __restrict__ Wf,
                     const float* __restrict__ bf,
                     float* __restrict__ out)
{
    // x as [j][l][k]: j*256 + l*4 + k -> phase-2 B fetch is an aligned float2 per lane.
    __shared__ float xbuf[NN * 256];               // 128 KB
    __shared__ float ebuf[2][NN * EB_LDA];         // 2 x 82.5 KB edge stream buffers
    // node tile lives in ebuf[1] during phases 0-1 (TDM only refills it at jb=0,
    // which is issued after the phase-1 barrier).

    const int b    = blockIdx.x;
    const int tid  = threadIdx.x;
    const int wave = tid >> 5;
    const int lane = tid & 31;
    const int n16  = lane & 15;
    const int hi   = lane >> 4;

    const float* eg = edge + (size_t)b * (NN * NN * EC);

#if HAVE_TDM
    // Kick the DMA for edge block 0 immediately; it streams in during phases 0-1.
    if (wave == 0)
        tdm_load_edge_block(eg, (unsigned)(uintptr_t)&ebuf[0][0]);
#endif

    // ---------------- Phase 0: stage node[b] (128x64) into LDS (ebuf[1]) --------
    float* nodebuf = &ebuf[1][0];
    {
        const float* ng  = node + (size_t)b * (NN * NIN_);
        const int    row = tid >> 3;
        const int    sub = tid & 7;
        #pragma unroll
        for (int c = 0; c < 8; ++c) {
            const int i = sub * 8 + c;
            nodebuf[row * NODE_LDA + i] = ng[row * NIN_ + i];
        }
    }
    __syncthreads();

    // ---------------- Phase 1: 5 GEMMs (128x64)@(64x64) via WMMA ----------------
    // Tile (jt, ot) per wave; mats 0..3 (We[e]) -> xbuf; mat 4 (Wf) stays in VGPRs.
    const int jt = wave & 7;          // j tile (== phase-2 i tile)
    const int ot = wave >> 3;         // o tile (== phase-2 l tile)
    const int o0 = ot * 16;
    v8f accS;                         // self_msg tile, reused as phase-2 seed
    {
        for (int mat = 0; mat < 4; ++mat) {
            const float* wp = We + mat * NIN_ * NOUT_;
            const float bias = be[mat * NOUT_ + o0 + n16];
            v8f acc;
            #pragma unroll
            for (int r = 0; r < 8; ++r) acc[r] = bias;

            #pragma unroll
            for (int i0 = 0; i0 < NIN_; i0 += 4) {
                v2f a = *(const v2f*)&nodebuf[(jt * 16 + n16) * NODE_LDA + i0 + 2 * hi];
                v2f bb;
                bb.x = wp[(i0 + 2 * hi    ) * NOUT_ + o0 + n16];
                bb.y = wp[(i0 + 2 * hi + 1) * NOUT_ + o0 + n16];
                acc = __builtin_amdgcn_wmma_f32_16x16x4_f32(
                        false, a, false, bb, (short)0, acc, false, false);
            }
            #pragma unroll
            for (int r = 0; r < 8; ++r)
                xbuf[(jt * 16 + r + 8 * hi) * 256 + (o0 + n16) * 4 + mat] = acc[r];
        }
        // self_msg = node @ Wf + bf : keep in registers (layout matches phase-2 C).
        const float bias = bf[o0 + n16];
        #pragma unroll
        for (int r = 0; r < 8; ++r) accS[r] = bias;
        #pragma unroll
        for (int i0 = 0; i0 < NIN_; i0 += 4) {
            v2f a = *(const v2f*)&nodebuf[(jt * 16 + n16) * NODE_LDA + i0 + 2 * hi];
            v2f bb;
            bb.x = Wf[(i0 + 2 * hi    ) * NOUT_ + o0 + n16];
            bb.y = Wf[(i0 + 2 * hi + 1) * NOUT_ + o0 + n16];
            accS = __builtin_amdgcn_wmma_f32_16x16x4_f32(
                    false, a, false, bb, (short)0, accS, false, false);
        }
    }
    __syncthreads();   // xbuf complete; nodebuf (ebuf[1]) free for TDM refill

    // ---------------- Phase 2: agg = self + sum_{j,k} e[i,j,k] * x[j,l,k] -------
    const int i0t = jt * 16;          // output row tile
    const int l0  = o0;               // output col tile
    v8f acc = accS;                   // fused self_msg add

    for (int jb = 0; jb < NJB; ++jb) {
        const int p = jb & 1;
#if HAVE_TDM
        if (wave == 0) {
            if (jb + 1 < NJB) {
                // Refill of buf[1-p] is safe: previous iteration's end barrier
                // guaranteed all waves finished reading it (jb=0: phase-1 barrier).
                tdm_load_edge_block(eg + (jb + 1) * (JB * EC),
                                    (unsigned)(uintptr_t)&ebuf[1 - p][0]);
                __builtin_amdgcn_s_wait_tensorcnt(1);  // in-order retire =>
            } else {                                   // current buffer complete
                __builtin_amdgcn_s_wait_tensorcnt(0);
            }
        }
        __syncthreads();
#else
        // Fallback: synchronous cooperative stage into the same padded layout.
        {
            const int row = tid >> 3, sub = tid & 7;
            const float* src = eg + row * (NN * EC) + jb * (JB * EC);
            float*       dst = ebuf[p] + row * EB_LDA;
            #pragma unroll
            for (int c = 0; c < 20; ++c) {
                const int w = sub * 20 + c;
                dst[w + (w >> 5)] = src[w];
            }
            if (jb + 1 < NJB) __builtin_prefetch(src + JB * EC, 0, 0);
        }
        __syncthreads();
#endif
        const float* eb = ebuf[p];
        #pragma unroll
        for (int j = 0; j < JB; ++j) {
            // A tile: rows i0..i0+15, K = channels 1+2hi, 2+2hi of column jb*JB+j.
            const int w0 = j * EC + 1 + 2 * hi;
            const int w1 = w0 + 1;
            v2f a;
            a.x = eb[(i0t + n16) * EB_LDA + w0 + (w0 >> 5)];
            a.y = eb[(i0t + n16) * EB_LDA + w1 + (w1 >> 5)];
            const int jg = jb * JB + j;
            v2f bb = *(const v2f*)&xbuf[jg * 256 + (l0 + n16) * 4 + 2 * hi];
            acc = __builtin_amdgcn_wmma_f32_16x16x4_f32(
                    false, a, false, bb, (short)0, acc, false, false);
        }
        __syncthreads();   // all waves done with buf[p] before it is refilled
    }

    // ---------------- Epilogue: relu + store ------------------------------------
    float* og = out + (size_t)b * (NN * NOUT_);
    #pragma unroll
    for (int r = 0; r < 8; ++r) {
        const float v = acc[r];
        og[(i0t + r + 8 * hi) * NOUT_ + l0 + n16] = v > 0.0f ? v : 0.0f;
    }
}

extern "C" void kernel_launch(void* const* d_in, const int* in_sizes, int n_in,
                              void* d_out, int out_size, void* d_ws, size_t ws_size,
                              hipStream_t stream) {
    (void)in_sizes; (void)n_in; (void)out_size; (void)d_ws; (void)ws_size;
    const float* node = (const float*)d_in[0];
    const float* edge = (const float*)d_in[1];
    const float* We   = (const float*)d_in[2];
    const float* be   = (const float*)d_in[3];
    const float* Wf   = (const float*)d_in[4];
    const float* bf   = (const float*)d_in[5];
    float* out = (float*)d_out;

    rgcn_fused_wmma<<<BN, 1024, 0, stream>>>(node, edge, We, be, Wf, bf, out);
}